// PathConv_32126355374223
// MI455X (gfx1250) — compile-verified
//
#include <hip/hip_runtime.h>

typedef __attribute__((ext_vector_type(2))) float v2f;
typedef __attribute__((ext_vector_type(4))) float v4f;
typedef __attribute__((ext_vector_type(8))) float v8f;
typedef __attribute__((ext_vector_type(4))) int   v4i;

constexpr int BB = 2;
constexpr int NN = 72;
constexpr int DD = 128;
constexpr int MM = BB * NN * NN;   // 10368 = 648*16 = 162*64 = 324*32
constexpr int LDS_STRIDE = DD + 4; // 132: bank-conflict-free (132 % 64 == 4)

#if defined(__has_builtin)
#if __has_builtin(__builtin_amdgcn_global_load_async_to_lds_b128) && \
    __has_builtin(__builtin_amdgcn_s_wait_asynccnt)
#define HAVE_ASYNC_LDS 1
#endif
#endif
#ifndef HAVE_ASYNC_LDS
#define HAVE_ASYNC_LDS 0
#endif

typedef __attribute__((address_space(1))) v4i* gptr_v4i;  // global (AS1)
typedef __attribute__((address_space(3))) v4i* lptr_v4i;  // LDS (AS3)

__device__ __forceinline__ void stage_chunk(const float* gsrc, float* lds_dst) {
#if HAVE_ASYNC_LDS
  __builtin_amdgcn_global_load_async_to_lds_b128(
      (gptr_v4i)gsrc, (lptr_v4i)lds_dst, 0, 0);
#else
  *(v4f*)lds_dst = *(const v4f*)gsrc;
#endif
}

__device__ __forceinline__ void stage_wait() {
#if HAVE_ASYNC_LDS
  __builtin_amdgcn_s_wait_asynccnt(0);
#endif
}

// ---------------------------------------------------------------------------
// Kernel 1: z[r,e] = path[r,:] @ W_hz + b_hz.  M-tile = 64 rows per block.
// A (64x128) async-staged in LDS; each wave computes 64x16 with 4 v8f
// accumulators, so every B (weight) fragment is reused 4x -> 4x less weight
// traffic and 4 WMMA per global B-pair in the inner loop.
// ---------------------------------------------------------------------------
__global__ __launch_bounds__(256)
void k_hz(const float* __restrict__ path, const float* __restrict__ Whz,
          const float* __restrict__ bhz, float* __restrict__ z) {
  __shared__ float lds_a[64][LDS_STRIDE];   // 33.8 KB

  const int tid  = threadIdx.x;
  const int lane = tid & 31;
  const int wave = tid >> 5;
  const int ml   = lane & 15;
  const int hi   = lane >> 4;
  const int r0   = blockIdx.x * 64;
  const int col  = wave * 16 + ml;

  // Stage A: 64 rows x 128 floats = 2048 x 16B chunks, 8 per thread.
#pragma unroll
  for (int c = tid; c < 2048; c += 256) {
    const int row = c >> 5;
    const int cc  = (c & 31) * 4;
    stage_chunk(path + (size_t)(r0 + row) * DD + cc, &lds_a[row][cc]);
  }
  stage_wait();
  __syncthreads();

  const float bias = bhz[col];
  v8f acc[4];
#pragma unroll
  for (int m = 0; m < 4; ++m) acc[m] = (v8f){};

#pragma unroll
  for (int k0 = 0; k0 < DD; k0 += 4) {
    const int k = k0 + 2 * hi;
    v2f b;
    b.x = Whz[(size_t)k * DD + col];
    b.y = Whz[(size_t)(k + 1) * DD + col];
#pragma unroll
    for (int m = 0; m < 4; ++m) {
      v2f a;
      a.x = lds_a[m * 16 + ml][k];
      a.y = lds_a[m * 16 + ml][k + 1];
      acc[m] = __builtin_amdgcn_wmma_f32_16x16x4_f32(false, a, false, b,
                                                     (short)0, acc[m],
                                                     false, false);
    }
  }
#pragma unroll
  for (int m = 0; m < 4; ++m)
#pragma unroll
    for (int v = 0; v < 8; ++v)
      z[(size_t)(r0 + m * 16 + v + 8 * hi) * DD + col] = acc[m][v] + bias;
}

// ---------------------------------------------------------------------------
// Kernel 2: tropical (max-plus) aggregation fused with max(z_conv, z).
// Thread owns 2 consecutive d-channels (v2f); 4x4 (i,j) tile per 64-thread
// block. All rows fully coalesced (64 lanes x 8B = one 512B z row).
// ---------------------------------------------------------------------------
__global__ __launch_bounds__(64)
void k_tropical(const float* __restrict__ zf, float* __restrict__ zmidf) {
  const v2f* __restrict__ Z = (const v2f*)zf;
  v2f* __restrict__ Zm      = (v2f*)zmidf;
  const int d2 = threadIdx.x;                 // 0..63 -> d = {2*d2, 2*d2+1}
  const int j0 = blockIdx.x * 4;
  const int i0 = blockIdx.y * 4;
  const int b  = blockIdx.z;
  const size_t base = (size_t)b * NN * NN * (DD / 2);

  v2f acc[4][4];
#pragma unroll
  for (int a = 0; a < 4; ++a)
#pragma unroll
    for (int c = 0; c < 4; ++c) {
      acc[a][c].x = -3.402823466e38f;
      acc[a][c].y = -3.402823466e38f;
    }

  for (int k = 0; k < NN; ++k) {
    v2f zi[4], zj[4];
#pragma unroll
    for (int t = 0; t < 4; ++t)
      zi[t] = Z[base + ((size_t)(i0 + t) * NN + k) * (DD / 2) + d2];
#pragma unroll
    for (int t = 0; t < 4; ++t)
      zj[t] = Z[base + ((size_t)k * NN + (j0 + t)) * (DD / 2) + d2];
#pragma unroll
    for (int a = 0; a < 4; ++a)
#pragma unroll
      for (int c = 0; c < 4; ++c) {
        const v2f s = zi[a] + zj[c];
        acc[a][c].x = fmaxf(acc[a][c].x, s.x);
        acc[a][c].y = fmaxf(acc[a][c].y, s.y);
      }
  }
#pragma unroll
  for (int a = 0; a < 4; ++a)
#pragma unroll
    for (int c = 0; c < 4; ++c) {
      const size_t idx =
          base + ((size_t)(i0 + a) * NN + (j0 + c)) * (DD / 2) + d2;
      const v2f zv = Z[idx];
      v2f r;
      r.x = fmaxf(acc[a][c].x, zv.x);
      r.y = fmaxf(acc[a][c].y, zv.y);
      Zm[idx] = r;
    }
}

// ---------------------------------------------------------------------------
// Kernel 3: out = relu(concat(n_i, zmid, n_j) @ W_zz + b_zz).  M-tile = 32
// rows per block (2 accumulators/wave -> each W_zz fragment reused 2x).
// K=384 as three 128-row segments; A tiles async-staged in LDS; out aliases
// zmid safely (block reads only its own rows before overwriting).
// ---------------------------------------------------------------------------
__global__ __launch_bounds__(256)
void k_zz(const float* __restrict__ node, const float* __restrict__ Wzz,
          const float* __restrict__ bzz, float* __restrict__ out) {
  __shared__ float lds_m[32][LDS_STRIDE];  // zmid rows (segment 1)
  __shared__ float lds_i[32][LDS_STRIDE];  // node_i rows (segment 0)
  __shared__ float lds_j[32][LDS_STRIDE];  // node_j rows (segment 2)

  const int tid  = threadIdx.x;
  const int lane = tid & 31;
  const int wave = tid >> 5;
  const int ml   = lane & 15;
  const int hi   = lane >> 4;
  const int r0   = blockIdx.x * 32;
  const int col  = wave * 16 + ml;

  // Stage all three A tiles: 3 x 1024 x 16B chunks, 12 per thread.
#pragma unroll
  for (int c = tid; c < 1024; c += 256) {
    const int row = c >> 5;
    const int cc  = (c & 31) * 4;
    const int r   = r0 + row;
    const int b   = r / (NN * NN);
    const int rem = r - b * NN * NN;
    const int i   = rem / NN;
    const int j   = rem - i * NN;
    stage_chunk(out + (size_t)r * DD + cc, &lds_m[row][cc]);
    stage_chunk(node + (size_t)(b * NN + i) * DD + cc, &lds_i[row][cc]);
    stage_chunk(node + (size_t)(b * NN + j) * DD + cc, &lds_j[row][cc]);
  }
  stage_wait();
  __syncthreads();

  const float bias = bzz[col];
  v8f acc[2];
  acc[0] = (v8f){};
  acc[1] = (v8f){};

  // Segment 0: n_i x W_zz[0:128,:]
#pragma unroll
  for (int k0 = 0; k0 < DD; k0 += 4) {
    const int k = k0 + 2 * hi;
    v2f w;
    w.x = Wzz[(size_t)k * DD + col];
    w.y = Wzz[(size_t)(k + 1) * DD + col];
#pragma unroll
    for (int m = 0; m < 2; ++m) {
      v2f a;
      a.x = lds_i[m * 16 + ml][k];
      a.y = lds_i[m * 16 + ml][k + 1];
      acc[m] = __builtin_amdgcn_wmma_f32_16x16x4_f32(false, a, false, w,
                                                     (short)0, acc[m],
                                                     false, false);
    }
  }
  // Segment 1: zmid x W_zz[128:256,:]
#pragma unroll
  for (int k0 = 0; k0 < DD; k0 += 4) {
    const int k = k0 + 2 * hi;
    v2f w;
    w.x = Wzz[(size_t)(DD + k) * DD + col];
    w.y = Wzz[(size_t)(DD + k + 1) * DD + col];
#pragma unroll
    for (int m = 0; m < 2; ++m) {
      v2f a;
      a.x = lds_m[m * 16 + ml][k];
      a.y = lds_m[m * 16 + ml][k + 1];
      acc[m] = __builtin_amdgcn_wmma_f32_16x16x4_f32(false, a, false, w,
                                                     (short)0, acc[m],
                                                     false, false);
    }
  }
  // Segment 2: n_j x W_zz[256:384,:]
#pragma unroll
  for (int k0 = 0; k0 < DD; k0 += 4) {
    const int k = k0 + 2 * hi;
    v2f w;
    w.x = Wzz[(size_t)(2 * DD + k) * DD + col];
    w.y = Wzz[(size_t)(2 * DD + k + 1) * DD + col];
#pragma unroll
    for (int m = 0; m < 2; ++m) {
      v2f a;
      a.x = lds_j[m * 16 + ml][k];
      a.y = lds_j[m * 16 + ml][k + 1];
      acc[m] = __builtin_amdgcn_wmma_f32_16x16x4_f32(false, a, false, w,
                                                     (short)0, acc[m],
                                                     false, false);
    }
  }

#pragma unroll
  for (int m = 0; m < 2; ++m)
#pragma unroll
    for (int v = 0; v < 8; ++v)
      out[(size_t)(r0 + m * 16 + v + 8 * hi) * DD + col] =
          fmaxf(acc[m][v] + bias, 0.0f);
}

// ---------------------------------------------------------------------------
extern "C" void kernel_launch(void* const* d_in, const int* in_sizes, int n_in,
                              void* d_out, int out_size, void* d_ws, size_t ws_size,
                              hipStream_t stream) {
  (void)in_sizes; (void)n_in; (void)out_size; (void)ws_size;
  const float* node = (const float*)d_in[0];  // (B*N, 128)
  const float* path = (const float*)d_in[1];  // (B, N, N, 128)
  const float* Whz  = (const float*)d_in[2];  // (128, 128)
  const float* bhz  = (const float*)d_in[3];  // (128,)
  const float* Wzz  = (const float*)d_in[4];  // (384, 128)
  const float* bzz  = (const float*)d_in[5];  // (128,)
  float* out = (float*)d_out;                 // (B, N, N, 128)
  float* z   = (float*)d_ws;                  // MM*128 floats = 5.31 MB scratch

  k_hz<<<MM / 64, 256, 0, stream>>>(path, Whz, bhz, z);
  k_tropical<<<dim3(NN / 4, NN / 4, BB), 64, 0, stream>>>(z, out);
  k_zz<<<MM / 32, 256, 0, stream>>>(node, Wzz, bzz, out);
}